// GNNLayer_8435315769871
// MI455X (gfx1250) — compile-verified
//
#include <hip/hip_runtime.h>
#include <stdint.h>

// GNN message-passing layer: h[v] = sum_{(u,v) in E} feat[u]
// N=100000 nodes, D=64 f32 features, E=1.2M edges (COO, random).
//
// MI455X strategy (memory-system bound; no dense matmul -> WMMA N/A):
//  - feat (25.6 MB) and out (25.6 MB) are L2-resident (192 MB L2)
//  - gather: 16 lanes/edge * float4 -> one coalesced 256B row per edge
//  - scatter: native non-returning global_atomic_add_f32 (relaxed, agent)
//  - edge indices: DOUBLE-BUFFERED async Global->LDS DMA
//    (global_load_async_to_lds_b32 + s_wait_asynccnt)   [CDNA5 path]
//  - global_prefetch_b8 hides gather latency

#define D_FEAT 64
constexpr int BLOCK = 256;   // 8 waves (wave32)
constexpr int CHUNK = 256;   // edges per staged chunk

__global__ __launch_bounds__(BLOCK)
void gnn_zero_kernel(float4* __restrict__ out, int n4) {
    int i = blockIdx.x * blockDim.x + threadIdx.x;
    if (i < n4) out[i] = float4{0.f, 0.f, 0.f, 0.f};
}

__device__ __forceinline__ void atomic_add_f32_agent(float* p, float v) {
    __hip_atomic_fetch_add(p, v, __ATOMIC_RELAXED, __HIP_MEMORY_SCOPE_AGENT);
}

// Low 32 bits of the LDS aperture flat address == wave-relative LDS byte
// offset (aperture base lives in bits [63:32]).
__device__ __forceinline__ uint32_t lds_addr_of(const void* p) {
    return (uint32_t)(uintptr_t)p;
}

// GVS form: global_mem_addr = SADDR(64) + VADDR(32, byte offset); LDS dest
// byte address in the vdst VGPR. Tracked on ASYNCcnt.
__device__ __forceinline__ void async_load_b32_to_lds(uint32_t lds_addr,
                                                      const int* gbase,
                                                      uint32_t byte_off) {
    asm volatile("global_load_async_to_lds_b32 %0, %1, %2"
                 :
                 : "v"(lds_addr), "v"(byte_off), "s"(gbase)
                 : "memory");
}

__global__ __launch_bounds__(BLOCK)
void gnn_scatter_kernel(const float* __restrict__ feat,
                        const int*   __restrict__ src,
                        const int*   __restrict__ dst,
                        float*       __restrict__ out,
                        int n_edges) {
    __shared__ int s_src[2][CHUNK];
    __shared__ int s_dst[2][CHUNK];

    const int tid   = threadIdx.x;
    const int lane4 = (tid & 15) * 4;   // feature offset in row: 0..60, step 4
    const int grp   = tid >> 4;         // 16 edge-groups of 16 lanes

    const int stride = gridDim.x * CHUNK;
    int base = blockIdx.x * CHUNK;
    if (base >= n_edges) return;        // uniform per block

    // Clamped issue => EXEC always full => every wave issues exactly 2 async
    // ops per staged chunk => deterministic ASYNCcnt for the <=2 / <=0 waits.
    auto stage = [&](int cbase, int buf) {
        int e = min(cbase + tid, n_edges - 1);
        uint32_t off = (uint32_t)e * 4u;
        async_load_b32_to_lds(lds_addr_of(&s_src[buf][tid]), src, off);
        async_load_b32_to_lds(lds_addr_of(&s_dst[buf][tid]), dst, off);
    };

    int cur = 0;
    stage(base, cur);                   // prologue: chunk 0 in flight

    for (; base < n_edges; base += stride) {
        const int next = base + stride;
        const bool have_next = (next < n_edges);   // uniform per block

        if (have_next) {
            stage(next, cur ^ 1);       // overlap: chunk i+1 DMA while we work
            asm volatile("s_wait_asynccnt 0x2" ::: "memory"); // chunk i landed
        } else {
            asm volatile("s_wait_asynccnt 0x0" ::: "memory");
        }
        __syncthreads();

        const int limit = min(CHUNK, n_edges - base);
        const int* __restrict__ csrc = s_src[cur];
        const int* __restrict__ cdst = s_dst[cur];

        #pragma unroll 4
        for (int it = 0; it < CHUNK / 16; ++it) {
            const int le = it * 16 + grp;

            // Prefetch next iteration's gather row into near caches.
            const int le_n = le + 16;
            if (le_n < limit) {
                __builtin_prefetch(feat + (size_t)csrc[le_n] * D_FEAT + lane4, 0, 0);
            }

            if (le < limit) {
                const int s = csrc[le];
                const int d = cdst[le];
                const float4 m =
                    *reinterpret_cast<const float4*>(feat + (size_t)s * D_FEAT + lane4);
                float* o = out + (size_t)d * D_FEAT + lane4;
                atomic_add_f32_agent(o + 0, m.x);
                atomic_add_f32_agent(o + 1, m.y);
                atomic_add_f32_agent(o + 2, m.z);
                atomic_add_f32_agent(o + 3, m.w);
            }
        }
        __syncthreads();                // all waves done reading buf[cur]
        cur ^= 1;                       // next iteration consumes the other buf
    }
}

extern "C" void kernel_launch(void* const* d_in, const int* in_sizes, int n_in,
                              void* d_out, int out_size, void* d_ws, size_t ws_size,
                              hipStream_t stream) {
    const float* feat = (const float*)d_in[0];
    const int*   src  = (const int*)d_in[1];
    const int*   dst  = (const int*)d_in[2];
    float*       out  = (float*)d_out;

    const int n_edges = in_sizes[1];

    // Zero the (poisoned) output before accumulation.
    const int n4 = out_size / 4;
    gnn_zero_kernel<<<(n4 + BLOCK - 1) / BLOCK, BLOCK, 0, stream>>>((float4*)out, n4);

    // ~4 chunks per block on average; grid-stride handles the rest.
    int nchunks = (n_edges + CHUNK - 1) / CHUNK;
    int nblocks = (nchunks + 3) / 4;
    if (nblocks < 1) nblocks = 1;
    gnn_scatter_kernel<<<nblocks, BLOCK, 0, stream>>>(feat, src, dst, out, n_edges);
}